// Attn_66271345377851
// MI455X (gfx1250) — compile-verified
//
#include <hip/hip_runtime.h>
#include <math.h>

#define HDIM 1024
#define BDIM 32
#define TDIM 2048

typedef __attribute__((ext_vector_type(2))) float v2f;
typedef __attribute__((ext_vector_type(4))) float v4f;
typedef __attribute__((ext_vector_type(8))) float v8f;

// ---------------------------------------------------------------------------
// Native CDNA5 tanh (V_TANH_F32, TRANS op — co-executes with VALU).
// Prefer the clang builtin (compiler inserts the TRANS hazard delay);
// fall back to inline asm + v_nop (ISA §7.4: 1 op after TRANS before use).
// ---------------------------------------------------------------------------
__device__ __forceinline__ float htanh(float x) {
#if __has_builtin(__builtin_amdgcn_tanhf)
    return __builtin_amdgcn_tanhf(x);
#else
    float r;
    asm volatile("v_tanh_f32 %0, %1\n\tv_nop" : "=v"(r) : "v"(x));
    return r;
#endif
}

// ---------------------------------------------------------------------------
// Kernel 1: q = relu(hidden @ Wq^T), M=32 N=1024 K=1024, f32 WMMA.
// One wave per 16x16 output tile. grid = (N/16, M/16), block = 32 (one wave32).
//
// V_WMMA_F32_16X16X4_F32 VGPR striping (ISA 7.12.2):
//   A 16x4 : lanes 0-15 -> M=lane, VGPR0=K0 VGPR1=K1; lanes 16-31 -> K2/K3
//   B 4x16 : lanes 0-15 -> N=lane, VGPR0=K0 VGPR1=K1; lanes 16-31 -> K2/K3
//   C/D    : VGPR v -> M = v (+8 for lanes 16-31), N = lane%16
// ---------------------------------------------------------------------------
__global__ __launch_bounds__(32) void q_gemm_wmma(const float* __restrict__ hidden,
                                                  const float* __restrict__ Wq,
                                                  float* __restrict__ q) {
    const int lane  = threadIdx.x;       // 0..31
    const int row   = lane & 15;
    const int hi    = lane >> 4;         // 0: K0/K1, 1: K2/K3
    const int ntile = blockIdx.x;        // 0..63
    const int mtile = blockIdx.y;        // 0..1
    const int m = mtile * 16 + row;      // A-row this lane feeds
    const int n = ntile * 16 + row;      // B-col this lane feeds (Wq row)

    v8f c = {};
    for (int k0 = 0; k0 < HDIM; k0 += 4) {
        const int kk = k0 + 2 * hi;
        v2f a = *(const v2f*)(hidden + (size_t)m * HDIM + kk);   // A[m][kk..kk+1]
        v2f b = *(const v2f*)(Wq     + (size_t)n * HDIM + kk);   // B[kk][n] = Wq[n][kk]
        c = __builtin_amdgcn_wmma_f32_16x16x4_f32(
                /*neg_a=*/false, a, /*neg_b=*/false, b,
                /*c_mod=*/(short)0, c, /*reuse_a=*/false, /*reuse_b=*/false);
    }

#pragma unroll
    for (int v = 0; v < 8; ++v) {
        const int mo = mtile * 16 + v + 8 * hi;
        float val = c[v];
        q[(size_t)mo * HDIM + ntile * 16 + row] = val > 0.f ? val : 0.f;  // fused ReLU
    }
}

// ---------------------------------------------------------------------------
// Kernel 2: fused content-dot + tanh-energy + mask. One wave per (b,t) row.
// Streams 512 MB with NT hints; q (128KB) and Wa (4KB) stay cache-resident.
// tanh runs on the TRANS pipe, hidden under the HBM stream (~22 us floor).
// ---------------------------------------------------------------------------
#define WAVES_PER_BLK 8
__global__ __launch_bounds__(256) void energy_kernel(const v4f* __restrict__ enc,   // [T,B,H/4]
                                                     const v4f* __restrict__ pl,    // [B,T,H/4]
                                                     const v4f* __restrict__ q,     // [B,H/4]
                                                     const v4f* __restrict__ wa,    // [H/4]
                                                     const int* __restrict__ mask,  // [B,T]
                                                     float* __restrict__ energies)  // [B,T]
{
    const int lane = threadIdx.x & 31;
    const int wave = threadIdx.x >> 5;
    const int r = blockIdx.x * WAVES_PER_BLK + wave;  // r = b*T + t
    const int b = r >> 11;                            // / TDIM
    const int t = r & (TDIM - 1);

    const v4f* __restrict__ erow = enc + ((size_t)t * BDIM + b) * (HDIM / 4);
    const v4f* __restrict__ prow = pl  + ((size_t)b * TDIM + t) * (HDIM / 4);
    const v4f* __restrict__ qrow = q   + (size_t)b * (HDIM / 4);

    // content = dot(enc row, q row)
    float acc = 0.f;
#pragma unroll
    for (int i = lane; i < HDIM / 4; i += 32) {
        v4f e  = __builtin_nontemporal_load(erow + i);
        v4f qq = qrow[i];
        acc += e.x * qq.x + e.y * qq.y + e.z * qq.z + e.w * qq.w;
    }
#pragma unroll
    for (int off = 16; off > 0; off >>= 1) acc += __shfl_xor(acc, off, 32);
    const float content = acc;   // all 32 lanes hold the sum (xor butterfly)

    // energy = sum_h tanh(content + pl[h]) * Wa[h]
    float acc2 = 0.f;
#pragma unroll
    for (int i = lane; i < HDIM / 4; i += 32) {
        v4f p = __builtin_nontemporal_load(prow + i);
        v4f w = wa[i];
        acc2 += htanh(content + p.x) * w.x + htanh(content + p.y) * w.y +
                htanh(content + p.z) * w.z + htanh(content + p.w) * w.w;
    }
#pragma unroll
    for (int off = 16; off > 0; off >>= 1) acc2 += __shfl_xor(acc2, off, 32);

    if (lane == 0)
        energies[r] = mask[r] ? -__builtin_inff() : acc2;
}

// ---------------------------------------------------------------------------
// Kernel 3: masked softmax over T per batch row. One block per b.
// ---------------------------------------------------------------------------
__global__ __launch_bounds__(256) void softmax_kernel(const float* __restrict__ energies, // [B,T]
                                                      float* __restrict__ attn)           // [B,T]
{
    __shared__ float sdata[256];
    __shared__ float sexp[TDIM];   // 8 KB of the 320 KB WGP LDS
    const int b   = blockIdx.x;
    const int tid = threadIdx.x;
    const float* __restrict__ row = energies + (size_t)b * TDIM;

    float m = -__builtin_inff();
    for (int t = tid; t < TDIM; t += 256) m = fmaxf(m, row[t]);
    sdata[tid] = m;
    __syncthreads();
    for (int s = 128; s > 0; s >>= 1) {
        if (tid < s) sdata[tid] = fmaxf(sdata[tid], sdata[tid + s]);
        __syncthreads();
    }
    const float rowmax = sdata[0];
    __syncthreads();   // protect sdata[0] read before reuse below

    float sum = 0.f;
    for (int t = tid; t < TDIM; t += 256) {
        float e = expf(row[t] - rowmax);   // exp(-inf - max) = 0 for masked slots
        sexp[t] = e;
        sum += e;
    }
    sdata[tid] = sum;
    __syncthreads();
    for (int s = 128; s > 0; s >>= 1) {
        if (tid < s) sdata[tid] += sdata[tid + s];
        __syncthreads();
    }
    const float inv = 1.f / sdata[0];

    for (int t = tid; t < TDIM; t += 256)
        attn[(size_t)b * TDIM + t] = sexp[t] * inv;
}

// ---------------------------------------------------------------------------
extern "C" void kernel_launch(void* const* d_in, const int* in_sizes, int n_in,
                              void* d_out, int out_size, void* d_ws, size_t ws_size,
                              hipStream_t stream) {
    const float* hidden = (const float*)d_in[0];  // [B,H]
    const float* enc    = (const float*)d_in[1];  // [T,B,H]
    const float* pl     = (const float*)d_in[2];  // [B,T,H]
    const int*   mask   = (const int*)  d_in[3];  // [B,T]
    const float* Wq     = (const float*)d_in[4];  // [H,H]
    const float* Wa     = (const float*)d_in[5];  // [1,H]

    float* attn     = (float*)d_out;                       // [B,1,T] -> B*T
    float* energies = (float*)d_out + (size_t)BDIM * TDIM; // [B,T]
    float* qws      = (float*)d_ws;                        // [B,H] = 128 KB scratch

    // 1) q = relu(hidden @ Wq^T) via f32 WMMA
    dim3 g1(HDIM / 16, BDIM / 16);
    q_gemm_wmma<<<g1, 32, 0, stream>>>(hidden, Wq, qws);

    // 2) fused content + tanh energy + mask  (memory-bound streamer)
    const int rows = BDIM * TDIM;
    energy_kernel<<<rows / WAVES_PER_BLK, 256, 0, stream>>>(
        (const v4f*)enc, (const v4f*)pl, (const v4f*)qws, (const v4f*)Wa,
        mask, energies);

    // 3) softmax over T per batch
    softmax_kernel<<<BDIM, 256, 0, stream>>>(energies, attn);
}